// SparseSpikingTemporalAttention_45354854646147
// MI455X (gfx1250) — compile-verified
//
#include <hip/hip_runtime.h>
#include <hip/hip_bf16.h>
#include <math.h>

typedef float v2f __attribute__((ext_vector_type(2)));
typedef float v8f __attribute__((ext_vector_type(8)));

#define T_DIM   8
#define N_NODES 20000
#define E_EDGES 160000
#define D_MODEL 64
#define HEADS   4
#define DHEAD   16
#define WP1     5
#define DPE     8
#define WK_STRIDE 72          // W_k is (64, 72) row-major
#define SCALE   0.25f         // 16^-0.5
#define NEGINF  (-1e30f)

// ---------- order-preserving float<->uint encoding for atomicMax ----------
__device__ __forceinline__ unsigned encf(float f) {
    unsigned u = __float_as_uint(f);
    return (u & 0x80000000u) ? ~u : (u | 0x80000000u);
}
__device__ __forceinline__ float decf(unsigned u) {
    return __uint_as_float((u & 0x80000000u) ? (u & 0x7fffffffu) : ~u);
}

// =========================================================================
// Kernel 1: fused Q / K_node / V projections via V_WMMA_F32_16X16X4_F32.
// One block = one 16-row tile of H (staged in LDS, pitch 65 to kill bank
// conflicts). 3 waves: wave0->Q, wave1->K_node, wave2->V. Each wave walks
// 4 column tiles x 16 K-steps of 16x16x4 fp32 WMMA.
// =========================================================================
__global__ __launch_bounds__(96) void proj_wmma_kernel(
    const float* __restrict__ H,
    const float* __restrict__ Wq,   // (64,64)
    const float* __restrict__ Wk,   // (64,72), use cols [0,64)
    const float* __restrict__ Wv,   // (64,64)
    float* __restrict__ Qo, float* __restrict__ Ko, float* __restrict__ Vo)
{
    __shared__ float hs[16 * 65];
    const int tid  = threadIdx.x;
    const size_t base = (size_t)blockIdx.x * (16 * 64);

    // coalesced stage of the 16x64 H tile
    for (int i = tid; i < 16 * 64; i += 96)
        hs[(i >> 6) * 65 + (i & 63)] = H[base + i];
    __syncthreads();

    const int wave = tid >> 5;
    const int lane = tid & 31;
    const int half = lane >> 4;   // lane group 0..15 / 16..31
    const int ln   = lane & 15;

    const float* W; int wstride; float* Out;
    if (wave == 0)      { W = Wq; wstride = D_MODEL;   Out = Qo; }
    else if (wave == 1) { W = Wk; wstride = WK_STRIDE; Out = Ko; }
    else                { W = Wv; wstride = D_MODEL;   Out = Vo; }

    for (int c0 = 0; c0 < 4; ++c0) {          // 16-wide output column tiles
        v8f acc = {};
        #pragma unroll
        for (int k0 = 0; k0 < 64; k0 += 4) {
            const int kk = k0 + 2 * half;
            v2f a, b;
            // A (16x4): lane half selects K pair {0,1} vs {2,3}
            a.x = hs[ln * 65 + kk];
            a.y = hs[ln * 65 + kk + 1];
            // B (4x16) = W^T tile: B[k][n] = W[c0*16+n][k]
            const float* wp = W + (size_t)(c0 * 16 + ln) * wstride + kk;
            b.x = wp[0];
            b.y = wp[1];
            acc = __builtin_amdgcn_wmma_f32_16x16x4_f32(
                false, a, false, b, (short)0, acc, false, false);
        }
        // C/D layout: VGPR j -> row j (lanes 0-15) / row j+8 (lanes 16-31)
        #pragma unroll
        for (int j = 0; j < 8; ++j) {
            const int row = j + half * 8;
            Out[base + (size_t)row * 64 + c0 * 16 + ln] = acc[j];
        }
    }
}

// =========================================================================
// Kernel 2: gate_log = log(clip(S,0,1) + 1e-6)
// =========================================================================
__global__ void gate_kernel(const float* __restrict__ S,
                            float* __restrict__ gate, int n)
{
    int i = blockIdx.x * blockDim.x + threadIdx.x;
    if (i < n) {
        float s = fminf(fmaxf(S[i], 0.f), 1.f);
        gate[i] = logf(s + 1e-6f);
    }
}

// =========================================================================
// Kernel 3: K_pe[w][d] = sum_p pe_table[w][p] * W_k[d][64+p]   (5x64)
// =========================================================================
__global__ void kpe_kernel(const float* __restrict__ pe,
                           const float* __restrict__ Wk,
                           float* __restrict__ Kpe)
{
    int i = threadIdx.x;
    if (i < WP1 * D_MODEL) {
        int w = i >> 6, d = i & 63;
        float s = 0.f;
        #pragma unroll
        for (int p = 0; p < DPE; ++p)
            s += pe[w * DPE + p] * Wk[(size_t)d * WK_STRIDE + 64 + p];
        Kpe[i] = s;
    }
}

// =========================================================================
// Kernel 4 (per t): zero output slice, zero denom, init maxbuf to -inf enc.
// =========================================================================
__global__ void init_kernel(float* __restrict__ out_t,
                            float* __restrict__ denom,
                            unsigned* __restrict__ maxbuf, int n)
{
    int i = blockIdx.x * blockDim.x + threadIdx.x;
    if (i < n) {
        out_t[i] = 0.f;
        if ((i & 15) == 0) {
            denom[i >> 4]  = 0.f;
            maxbuf[i >> 4] = encf(NEGINF);
        }
    }
}

// =========================================================================
// Kernel 5 (per t): per (edge, head) scores for all valid w, store to ws,
// register-max over w, single atomicMax per (edge,head) into dst segment.
// =========================================================================
__global__ __launch_bounds__(256) void score_kernel(
    const float* __restrict__ Q, const float* __restrict__ Kn,
    const float* __restrict__ Kpe, const float* __restrict__ gate,
    const float* __restrict__ rel_bias, const int* __restrict__ edge,
    float* __restrict__ scores, unsigned* __restrict__ maxbuf, int t)
{
    int gid = blockIdx.x * blockDim.x + threadIdx.x;
    if (gid >= E_EDGES * HEADS) return;
    const int e = gid >> 2, h = gid & 3;
    const int se = edge[e];
    const int de = edge[E_EDGES + e];

    const float4* qp = (const float4*)(Q + ((size_t)t * N_NODES + de) * 64 + h * DHEAD);
    const float4 q0 = qp[0], q1 = qp[1], q2 = qp[2], q3 = qp[3];

    float smax = NEGINF;
    for (int w = 0; w < WP1 && w <= t; ++w) {
        const int tp = t - w;
        const float4* kp = (const float4*)(Kn + ((size_t)tp * N_NODES + se) * 64 + h * DHEAD);
        const float4* pp = (const float4*)(Kpe + w * 64 + h * DHEAD);
        float4 k0 = kp[0], k1 = kp[1], k2 = kp[2], k3 = kp[3];
        float4 p0 = pp[0], p1 = pp[1], p2 = pp[2], p3 = pp[3];
        float s =
            (k0.x + p0.x) * q0.x + (k0.y + p0.y) * q0.y + (k0.z + p0.z) * q0.z + (k0.w + p0.w) * q0.w +
            (k1.x + p1.x) * q1.x + (k1.y + p1.y) * q1.y + (k1.z + p1.z) * q1.z + (k1.w + p1.w) * q1.w +
            (k2.x + p2.x) * q2.x + (k2.y + p2.y) * q2.y + (k2.z + p2.z) * q2.z + (k2.w + p2.w) * q2.w +
            (k3.x + p3.x) * q3.x + (k3.y + p3.y) * q3.y + (k3.z + p3.z) * q3.z + (k3.w + p3.w) * q3.w;
        s = s * SCALE + rel_bias[w] + gate[(size_t)tp * N_NODES + se];
        scores[((size_t)w * E_EDGES + e) * HEADS + h] = s;
        smax = fmaxf(smax, s);
    }
    atomicMax(maxbuf + (size_t)de * HEADS + h, encf(smax));
}

// =========================================================================
// Kernel 6 (per t): ex = exp(score - max), reduce over w in registers, then
// one denom atomic + 16 numer atomics per (edge,head) straight into d_out.
// =========================================================================
__global__ __launch_bounds__(256) void accum_kernel(
    const float* __restrict__ V, const float* __restrict__ scores,
    const int* __restrict__ edge, const unsigned* __restrict__ maxbuf,
    float* __restrict__ denom, float* __restrict__ out_t, int t)
{
    int gid = blockIdx.x * blockDim.x + threadIdx.x;
    if (gid >= E_EDGES * HEADS) return;
    const int e = gid >> 2, h = gid & 3;
    const int se = edge[e];
    const int de = edge[E_EDGES + e];

    const float maxv = decf(maxbuf[(size_t)de * HEADS + h]);
    float exsum = 0.f;
    float acc[DHEAD];
    #pragma unroll
    for (int c = 0; c < DHEAD; ++c) acc[c] = 0.f;

    for (int w = 0; w < WP1 && w <= t; ++w) {
        const float ex = expf(scores[((size_t)w * E_EDGES + e) * HEADS + h] - maxv);
        exsum += ex;
        const float4* vp = (const float4*)(V + ((size_t)(t - w) * N_NODES + se) * 64 + h * DHEAD);
        #pragma unroll
        for (int j = 0; j < 4; ++j) {
            float4 v = vp[j];
            acc[j * 4 + 0] += ex * v.x;
            acc[j * 4 + 1] += ex * v.y;
            acc[j * 4 + 2] += ex * v.z;
            acc[j * 4 + 3] += ex * v.w;
        }
    }
    atomicAdd(denom + (size_t)de * HEADS + h, exsum);
    float* op = out_t + (size_t)de * 64 + h * DHEAD;
    #pragma unroll
    for (int c = 0; c < DHEAD; ++c) atomicAdd(op + c, acc[c]);
}

// =========================================================================
// Kernel 7 (per t): normalize by max(denom, 1e-12)
// =========================================================================
__global__ void norm_kernel(float* __restrict__ out_t,
                            const float* __restrict__ denom, int n)
{
    int i = blockIdx.x * blockDim.x + threadIdx.x;
    if (i < n) {
        const int node = i >> 6;
        const int h = (i >> 4) & 3;
        out_t[i] /= fmaxf(denom[node * 4 + h], 1e-12f);
    }
}

// =========================================================================
extern "C" void kernel_launch(void* const* d_in, const int* in_sizes, int n_in,
                              void* d_out, int out_size, void* d_ws, size_t ws_size,
                              hipStream_t stream)
{
    const float* H    = (const float*)d_in[0];   // (8, 20000, 64)
    const float* S    = (const float*)d_in[1];   // (8, 20000)
    const float* Wq   = (const float*)d_in[2];   // (64, 64)
    const float* Wk   = (const float*)d_in[3];   // (64, 72)
    const float* Wv   = (const float*)d_in[4];   // (64, 64)
    const float* pe   = (const float*)d_in[5];   // (5, 8)
    const float* rb   = (const float*)d_in[6];   // (5,)
    const int*   edge = (const int*)d_in[7];     // (2, 160000)
    float* out = (float*)d_out;                  // (8, 20000, 64)

    // workspace layout (floats)
    const size_t TN64 = (size_t)T_DIM * N_NODES * 64;   // 10,240,000
    float* ws      = (float*)d_ws;
    float* Qb      = ws;
    float* Kb      = ws + TN64;
    float* Vb      = ws + 2 * TN64;
    float* gateb   = ws + 3 * TN64;
    float* Kpeb    = gateb + (size_t)T_DIM * N_NODES;
    float* scoresb = Kpeb + WP1 * 64;
    float* denomb  = scoresb + (size_t)WP1 * E_EDGES * HEADS;
    unsigned* maxb = (unsigned*)(denomb + (size_t)N_NODES * HEADS);

    // 1) projections (WMMA fp32)
    proj_wmma_kernel<<<(T_DIM * N_NODES) / 16, 96, 0, stream>>>(
        H, Wq, Wk, Wv, Qb, Kb, Vb);

    // 2) gate log
    {
        int n = T_DIM * N_NODES;
        gate_kernel<<<(n + 255) / 256, 256, 0, stream>>>(S, gateb, n);
    }

    // 3) PE contribution to K
    kpe_kernel<<<1, 384, 0, stream>>>(pe, Wk, Kpeb);

    // 4) per-timestep attention pipeline
    const int nEH  = E_EDGES * HEADS;       // 640,000
    const int nOut = N_NODES * 64;          // 1,280,000
    for (int t = 0; t < T_DIM; ++t) {
        float* out_t = out + (size_t)t * nOut;
        init_kernel<<<(nOut + 255) / 256, 256, 0, stream>>>(out_t, denomb, maxb, nOut);
        score_kernel<<<(nEH + 255) / 256, 256, 0, stream>>>(
            Qb, Kb, Kpeb, gateb, rb, edge, scoresb, maxb, t);
        accum_kernel<<<(nEH + 255) / 256, 256, 0, stream>>>(
            Vb, scoresb, edge, maxb, denomb, out_t, t);
        norm_kernel<<<(nOut + 255) / 256, 256, 0, stream>>>(out_t, denomb, nOut);
    }
    (void)in_sizes; (void)n_in; (void)out_size; (void)ws_size;
}